// BipartiteCommitteeGAT_48344151884193
// MI455X (gfx1250) — compile-verified
//
#include <hip/hip_runtime.h>
#include <math.h>

#define N_POL 1000
#define N_TICK 98000
#define N_COMM 1000
#define NN (N_POL + N_TICK + N_COMM)   /* 100000 */
#define EE 1000000
#define E2 (EE + NN)                   /* 1100000 (edges + self loops) */
#define EMB 32
#define HID 64
#define OUTD 32
#define HEADS 4
#define F1 (HID / HEADS)               /* 16 */

typedef __attribute__((ext_vector_type(2))) float v2f;
typedef __attribute__((ext_vector_type(8))) float v8f;
typedef __attribute__((ext_vector_type(4))) unsigned int v4u;
typedef __attribute__((ext_vector_type(8))) int v8i;
typedef __attribute__((ext_vector_type(4))) int v4i;

// ---------- helpers ----------
__device__ __forceinline__ int f2mono(float f) {
  int i = __float_as_int(f);
  return (i >= 0) ? i : (i ^ 0x7fffffff);   // monotonic int encoding of float order
}
__device__ __forceinline__ float mono2f(int m) {
  return __int_as_float((m >= 0) ? m : (m ^ 0x7fffffff));
}

// ---------- init accumulators ----------
__global__ void k_init(float* cnt, float* loopa, int* mx1, float* sm1, float* out1,
                       int* mx2, float* sm2, float* out2) {
  int stride = gridDim.x * blockDim.x;
  for (int i = blockIdx.x * blockDim.x + threadIdx.x; i < NN * HID; i += stride) {
    out1[i] = 0.f;
    if (i < NN * OUTD) out2[i] = 0.f;
    if (i < NN * 5)    loopa[i] = 0.f;
    if (i < NN * HEADS) { mx1[i] = (int)0x80000000; sm1[i] = 0.f; }
    if (i < NN)        { cnt[i] = 0.f; mx2[i] = (int)0x80000000; sm2[i] = 0.f; }
  }
}

// ---------- fused node features + LayerNorm: one wave(32) per node, lane = feature ----------
__global__ void k_node_feat(const float* __restrict__ pol_feat, const int* __restrict__ state_ids,
                            const int* __restrict__ sector_ids, const int* __restrict__ industry_ids,
                            const float* __restrict__ comp_scalar,
                            const float* __restrict__ pol_W, const float* __restrict__ pol_b,
                            const float* __restrict__ state_E, const float* __restrict__ sector_E,
                            const float* __restrict__ industry_E, const float* __restrict__ comp_W,
                            const float* __restrict__ comp_b, const float* __restrict__ comm_E,
                            const float* __restrict__ ln_g, const float* __restrict__ ln_b,
                            float* __restrict__ x1) {
  int node = blockIdx.x * (blockDim.x >> 5) + (threadIdx.x >> 5);
  int j = threadIdx.x & 31;
  if (node >= NN) return;
  float v;
  if (node < N_POL) {
    float s = pol_b[j];
#pragma unroll
    for (int k = 0; k < 7; ++k) s += pol_feat[node * 7 + k] * pol_W[k * EMB + j];
    v = fmaxf(s, 0.f) + state_E[state_ids[node] * EMB + j];
  } else if (node < N_POL + N_TICK) {
    int t = node - N_POL;
    float s = comp_b[j];
    int sec = sector_ids[t], ind = industry_ids[t];
#pragma unroll
    for (int k = 0; k < 8; ++k) s += sector_E[sec * 8 + k] * comp_W[k * EMB + j];
#pragma unroll
    for (int k = 0; k < 8; ++k) s += industry_E[ind * 8 + k] * comp_W[(8 + k) * EMB + j];
    s += comp_scalar[t] * comp_W[16 * EMB + j];
    v = fmaxf(s, 0.f);
  } else {
    v = comm_E[(node - N_POL - N_TICK) * EMB + j];
  }
  // LayerNorm across the 32 lanes of this wave (wave32)
  float sum = v;
#pragma unroll
  for (int m = 16; m >= 1; m >>= 1) sum += __shfl_xor(sum, m, 32);
  float mu = sum * (1.f / 32.f);
  float d = v - mu;
  float vs = d * d;
#pragma unroll
  for (int m = 16; m >= 1; m >>= 1) vs += __shfl_xor(vs, m, 32);
  float var = vs * (1.f / 32.f);
  x1[node * EMB + j] = d * rsqrtf(var + 1e-5f) * ln_g[j] + ln_b[j];
}

// ---------- per-dst degree + edge_attr sum (for self-loop fill 'mean') ----------
__global__ void k_deg(const int* __restrict__ ei, const float* __restrict__ eattr,
                      float* cnt, float* loopa) {
  int stride = gridDim.x * blockDim.x;
  for (int e = blockIdx.x * blockDim.x + threadIdx.x; e < EE; e += stride) {
    int dst = ei[EE + e];
    atomicAdd(&cnt[dst], 1.f);
#pragma unroll
    for (int k = 0; k < 5; ++k) atomicAdd(&loopa[dst * 5 + k], eattr[e * 5 + k]);
  }
}
__global__ void k_loop_div(float* loopa, const float* __restrict__ cnt) {
  int i = blockIdx.x * blockDim.x + threadIdx.x;
  if (i < NN * 5) loopa[i] /= fmaxf(cnt[i / 5], 1.f);
}

// ---------- collapse (eattr@eW)*a_e into a 5xH vector per layer ----------
__global__ void k_we(const float* __restrict__ c1_eW, const float* __restrict__ c1_ae,
                     const float* __restrict__ c2_eW, const float* __restrict__ c2_ae,
                     float* We1, float* We2) {
  int t = threadIdx.x;
  if (t < 20) {
    int k = t / HEADS, h = t % HEADS;
    float s = 0.f;
    for (int f = 0; f < F1; ++f) s += c1_eW[k * HID + h * F1 + f] * c1_ae[h * F1 + f];
    We1[k * HEADS + h] = s;
  } else if (t < 25) {
    int k = t - 20;
    float s = 0.f;
    for (int f = 0; f < OUTD; ++f) s += c2_eW[k * OUTD + f] * c2_ae[f];
    We2[k] = s;
  }
}

// ---------- f32 GEMM via V_WMMA_F32_16X16X4_F32 ----------
// B (K x Nn weight matrix, <=16KB) is staged into LDS once per workgroup with the
// Tensor Data Mover; WMMA B-fragments then come from LDS (ds_load).
__global__ void k_gemm_wmma(const float* __restrict__ A, const float* __restrict__ B,
                            float* __restrict__ C, int Nn, int K, int totalTiles) {
  __shared__ float Bs[HID * HID];     // max 64x64 f32 = 16KB of the 320KB/WGP LDS

#if __has_builtin(__builtin_amdgcn_tensor_load_to_lds)
  if ((threadIdx.x >> 5) == 0) {      // wave 0 of the block drives the TDM
    unsigned ldsoff = (unsigned)(unsigned long long)(uintptr_t)Bs;  // LDS byte offset
    unsigned long long ga = (unsigned long long)(uintptr_t)B;
    // D# group0: count=1 | lds_addr | global_addr | type=2
    v4u g0 = { 1u, ldsoff, (unsigned)(ga & 0xffffffffu),
               (unsigned)((ga >> 32) & 0x01ffffffu) | 0x80000000u };
    // D# group1: data_size=4B; tensor_dim0=Nn; tensor_dim1=K; tile_dim0=Nn;
    //            tile_dim1=K; tensor_dim0_stride=Nn (contiguous 2D tile == tensor)
    v8i g1 = { (int)0x00020000,
               (int)((unsigned)(Nn & 0xffff) << 16),
               (int)((unsigned)(K & 0xffff) << 16),
               (int)((unsigned)(Nn & 0xffff) << 16),
               (int)(K & 0xffff),
               (int)Nn,
               0, 0 };
    v4i g2 = { 0, 0, 0, 0 };
    v4i g3 = { 0, 0, 0, 0 };
#if __clang_major__ >= 23
    v8i g4 = { 0, 0, 0, 0, 0, 0, 0, 0 };
    __builtin_amdgcn_tensor_load_to_lds(g0, g1, g2, g3, g4, 0);
#else
    __builtin_amdgcn_tensor_load_to_lds(g0, g1, g2, g3, 0);
#endif
    __builtin_amdgcn_s_wait_tensorcnt(0);
  }
#else
  for (int i = threadIdx.x; i < K * Nn; i += blockDim.x) Bs[i] = B[i];
#endif
  __syncthreads();                    // all waves reach this before any early exit

  int wave = (blockIdx.x * blockDim.x + threadIdx.x) >> 5;  // wave-uniform
  if (wave >= totalTiles) return;                            // whole wave exits together
  int tilesN = Nn >> 4;
  int tm = (wave / tilesN) << 4;
  int tn = (wave % tilesN) << 4;
  int lane = threadIdx.x & 31;
  int half = lane >> 4;               // 0: K pair {0,1}, 1: K pair {2,3}
  int l15 = lane & 15;
  const float* Arow = A + (tm + l15) * K;
  v8f acc = {};
  for (int k0 = 0; k0 < K; k0 += 4) {
    v2f a, b;
    // A 16x4 layout: lanes0-15 -> K=k0,k0+1 ; lanes16-31 -> K=k0+2,k0+3 (M = lane&15)
    a.x = Arow[k0 + 2 * half];
    a.y = Arow[k0 + 2 * half + 1];
    // B 4x16 fragment from LDS, mirrored K split
    b.x = Bs[(k0 + 2 * half) * Nn + tn + l15];
    b.y = Bs[(k0 + 2 * half + 1) * Nn + tn + l15];
    acc = __builtin_amdgcn_wmma_f32_16x16x4_f32(false, a, false, b, (short)0, acc,
                                                false, false);
  }
#pragma unroll
  for (int v = 0; v < 8; ++v) {
    // D layout: VGPR v -> M = v (lanes 0-15) / v+8 (lanes 16-31), N = lane&15
    C[(tm + v + 8 * half) * Nn + tn + l15] = acc[v];
  }
}

// ---------- per-node attention terms ----------
__global__ void k_node_alpha(const float* __restrict__ h, const float* __restrict__ a_s,
                             const float* __restrict__ a_d, float* asrc, float* adst,
                             int H, int F) {
  int i = blockIdx.x * blockDim.x + threadIdx.x;
  if (i >= NN * H) return;
  int n = i / H, hd = i % H;
  const float* hp = h + n * H * F + hd * F;
  float ss = 0.f, sd = 0.f;
  for (int f = 0; f < F; ++f) { ss += hp[f] * a_s[hd * F + f]; sd += hp[f] * a_d[hd * F + f]; }
  asrc[i] = ss;
  adst[i] = sd;
}

__device__ __forceinline__ void fetch_edge(int e, const int* __restrict__ ei,
                                           const float* __restrict__ eattr,
                                           const float* __restrict__ loopa,
                                           int& src, int& dst, float* ea) {
  if (e < EE) {
    src = ei[e]; dst = ei[EE + e];
#pragma unroll
    for (int k = 0; k < 5; ++k) ea[k] = eattr[e * 5 + k];
  } else {
    int n = e - EE; src = n; dst = n;
#pragma unroll
    for (int k = 0; k < 5; ++k) ea[k] = loopa[n * 5 + k];
  }
}

// ---------- edge logits + per-dst running max (monotonic-int atomic max) ----------
__global__ void k_edge_logit(const int* __restrict__ ei, const float* __restrict__ eattr,
                             const float* __restrict__ loopa, const float* __restrict__ We,
                             const float* __restrict__ asrc, const float* __restrict__ adst,
                             float* elog, int* mx, int H) {
  int stride = gridDim.x * blockDim.x;
  for (int e = blockIdx.x * blockDim.x + threadIdx.x; e < E2; e += stride) {
    int src, dst; float ea[5];
    fetch_edge(e, ei, eattr, loopa, src, dst, ea);
    for (int hd = 0; hd < H; ++hd) {
      float l = asrc[src * H + hd] + adst[dst * H + hd];
#pragma unroll
      for (int k = 0; k < 5; ++k) l += ea[k] * We[k * H + hd];
      l = (l > 0.f) ? l : 0.2f * l;           // leaky_relu(0.2)
      elog[e * H + hd] = l;
      atomicMax(&mx[dst * H + hd], f2mono(l));
    }
  }
}

// ---------- exp(logit - max) + per-dst sum ----------
__global__ void k_edge_exp(const int* __restrict__ ei, float* elog, const int* __restrict__ mx,
                           float* sm, int H) {
  int stride = gridDim.x * blockDim.x;
  for (int e = blockIdx.x * blockDim.x + threadIdx.x; e < E2; e += stride) {
    int dst = (e < EE) ? ei[EE + e] : (e - EE);
    for (int hd = 0; hd < H; ++hd) {
      float m = mono2f(mx[dst * H + hd]);
      float ex = expf(elog[e * H + hd] - m);
      elog[e * H + hd] = ex;
      atomicAdd(&sm[dst * H + hd], ex);
    }
  }
}

// ---------- weighted scatter: out[dst] += alpha * h[src] ----------
// One lane per OUTPUT ELEMENT: consecutive lanes cover the contiguous H*F slice of
// one edge -> coalesced h[src] reads and same-cacheline atomic adds at the L2.
__global__ void k_edge_scatter(const int* __restrict__ ei, const float* __restrict__ elog,
                               const float* __restrict__ sm, const float* __restrict__ h,
                               float* out, int H, int HFlog2, int Flog2) {
  int HFm = (1 << HFlog2) - 1;
  int total = E2 << HFlog2;            // <= 70.4M, fits in int
  int stride = gridDim.x * blockDim.x;
  for (int i = blockIdx.x * blockDim.x + threadIdx.x; i < total; i += stride) {
    int e = i >> HFlog2;               // shared by (1<<HFlog2) consecutive lanes
    int r = i & HFm;                   // hd*F + f
    int hd = r >> Flog2;
    int src, dst;
    if (e < EE) { src = ei[e]; dst = ei[EE + e]; } else { src = dst = e - EE; }
    float alpha = elog[e * H + hd] / sm[dst * H + hd];
    atomicAdd(&out[(dst << HFlog2) + r], h[(src << HFlog2) + r] * alpha);
  }
}

// ---------- bias (+ optional ELU), in place ----------
__global__ void k_bias_elu(float* x, const float* __restrict__ b, int D, int do_elu) {
  int total = NN * D;
  int stride = gridDim.x * blockDim.x;
  for (int i = blockIdx.x * blockDim.x + threadIdx.x; i < total; i += stride) {
    float v = x[i] + b[i % D];
    if (do_elu) v = (v > 0.f) ? v : (expf(v) - 1.f);
    x[i] = v;
  }
}

extern "C" void kernel_launch(void* const* d_in, const int* in_sizes, int n_in,
                              void* d_out, int out_size, void* d_ws, size_t ws_size,
                              hipStream_t stream) {
  (void)in_sizes; (void)n_in; (void)out_size; (void)ws_size;
  const int*   edge_index  = (const int*)d_in[0];
  const float* edge_attr   = (const float*)d_in[1];
  const float* pol_feat    = (const float*)d_in[2];
  const int*   state_ids   = (const int*)d_in[3];
  const int*   sector_ids  = (const int*)d_in[4];
  const int*   industry_ids= (const int*)d_in[5];
  const float* comp_scalar = (const float*)d_in[6];
  const float* pol_W  = (const float*)d_in[7];
  const float* pol_b  = (const float*)d_in[8];
  const float* state_E= (const float*)d_in[9];
  const float* sector_E   = (const float*)d_in[10];
  const float* industry_E = (const float*)d_in[11];
  const float* comp_W = (const float*)d_in[12];
  const float* comp_b = (const float*)d_in[13];
  const float* comm_E = (const float*)d_in[14];
  const float* ln_g = (const float*)d_in[15];
  const float* ln_b = (const float*)d_in[16];
  const float* c1_W = (const float*)d_in[17];
  const float* c1_as= (const float*)d_in[18];
  const float* c1_ad= (const float*)d_in[19];
  const float* c1_eW= (const float*)d_in[20];
  const float* c1_ae= (const float*)d_in[21];
  const float* c1_b = (const float*)d_in[22];
  const float* c2_W = (const float*)d_in[23];
  const float* c2_as= (const float*)d_in[24];
  const float* c2_ad= (const float*)d_in[25];
  const float* c2_eW= (const float*)d_in[26];
  const float* c2_ae= (const float*)d_in[27];
  const float* c2_b = (const float*)d_in[28];
  float* out = (float*)d_out;

  float* ws = (float*)d_ws;
  size_t off = 0;
  auto alloc = [&](size_t n) { float* p = ws + off; off += n; return p; };
  float* x1    = alloc((size_t)NN * EMB);
  float* cnt   = alloc(NN);
  float* loopa = alloc((size_t)NN * 5);
  float* We1   = alloc(32);
  float* We2   = alloc(32);
  float* h1    = alloc((size_t)NN * HID);
  float* asrc1 = alloc((size_t)NN * HEADS);
  float* adst1 = alloc((size_t)NN * HEADS);
  float* elog1 = alloc((size_t)E2 * HEADS);
  int*   mx1   = (int*)alloc((size_t)NN * HEADS);
  float* sm1   = alloc((size_t)NN * HEADS);
  float* out1  = alloc((size_t)NN * HID);    // becomes x2 in-place after bias+ELU
  float* h2    = alloc((size_t)NN * OUTD);
  float* asrc2 = alloc(NN);
  float* adst2 = alloc(NN);
  float* elog2 = alloc(E2);
  int*   mx2   = (int*)alloc(NN);
  float* sm2   = alloc(NN);

  const int TB = 256;
  k_init<<<4096, TB, 0, stream>>>(cnt, loopa, mx1, sm1, out1, mx2, sm2, out);
  k_node_feat<<<(NN + 7) / 8, TB, 0, stream>>>(pol_feat, state_ids, sector_ids, industry_ids,
                                               comp_scalar, pol_W, pol_b, state_E, sector_E,
                                               industry_E, comp_W, comp_b, comm_E, ln_g, ln_b, x1);
  k_deg<<<4096, TB, 0, stream>>>(edge_index, edge_attr, cnt, loopa);
  k_loop_div<<<(NN * 5 + TB - 1) / TB, TB, 0, stream>>>(loopa, cnt);
  k_we<<<1, 32, 0, stream>>>(c1_eW, c1_ae, c2_eW, c2_ae, We1, We2);

  // ---- layer 1 ----
  int tiles1 = (NN / 16) * (HID / 16);                 // 25000 tiles, 8 waves/block
  k_gemm_wmma<<<(tiles1 + 7) / 8, TB, 0, stream>>>(x1, c1_W, h1, HID, EMB, tiles1);
  k_node_alpha<<<(NN * HEADS + TB - 1) / TB, TB, 0, stream>>>(h1, c1_as, c1_ad, asrc1, adst1,
                                                              HEADS, F1);
  k_edge_logit<<<4096, TB, 0, stream>>>(edge_index, edge_attr, loopa, We1, asrc1, adst1,
                                        elog1, mx1, HEADS);
  k_edge_exp<<<4096, TB, 0, stream>>>(edge_index, elog1, mx1, sm1, HEADS);
  k_edge_scatter<<<16384, TB, 0, stream>>>(edge_index, elog1, sm1, h1, out1,
                                           HEADS, /*HFlog2=*/6, /*Flog2=*/4);
  k_bias_elu<<<4096, TB, 0, stream>>>(out1, c1_b, HID, 1);

  // ---- layer 2 ----
  int tiles2 = (NN / 16) * (OUTD / 16);                // 12500 tiles
  k_gemm_wmma<<<(tiles2 + 7) / 8, TB, 0, stream>>>(out1, c2_W, h2, OUTD, HID, tiles2);
  k_node_alpha<<<(NN + TB - 1) / TB, TB, 0, stream>>>(h2, c2_as, c2_ad, asrc2, adst2, 1, OUTD);
  k_edge_logit<<<4096, TB, 0, stream>>>(edge_index, edge_attr, loopa, We2, asrc2, adst2,
                                        elog2, mx2, 1);
  k_edge_exp<<<4096, TB, 0, stream>>>(edge_index, elog2, mx2, sm2, 1);
  k_edge_scatter<<<16384, TB, 0, stream>>>(edge_index, elog2, sm2, h2, out,
                                           1, /*HFlog2=*/5, /*Flog2=*/5);
  k_bias_elu<<<4096, TB, 0, stream>>>(out, c2_b, OUTD, 0);
}